// ImprovedGNN_91130616087114
// MI455X (gfx1250) — compile-verified
//
#include <hip/hip_runtime.h>

// 3-layer GCN for MI455X (gfx1250, wave32).
// GEMMs via V_WMMA_F32_16X16X4_F32 (fp32-exact); edge scatter via fp32 global
// atomics (L2-resident: node features are 25.6MB << 192MB L2); BatchNorm via
// two-level reduction. HBM-resident traffic is only the edge lists.
//
// v2 changes:
//  - W staged in LDS as K-pairs (float2), so each B fragment is ONE aligned
//    ds_load_b64 (no v_dual_mov repacking). float2 row stride 80 => row+1 is
//    +32 banks: lanes 0-15 and 16-31 land on disjoint bank halves.
//  - elementwise kernels (init_agg, bn) vectorized to float4 b128 traffic.

#define BN_EPS 1e-5f
#define LDS_P_STRIDE 80   // float2 units per k-pair row (64 + 16 pad)

typedef float v2f __attribute__((ext_vector_type(2)));
typedef float v4f __attribute__((ext_vector_type(4)));
typedef float v8f __attribute__((ext_vector_type(8)));

// ---------------- GEMM: H = X @ W + b   (X: nrows x 64, W: 64 x 64) --------
__global__ __launch_bounds__(256) void gcn_gemm_wmma(
    const float* __restrict__ X, const float* __restrict__ W,
    const float* __restrict__ bias, float* __restrict__ Hout, int nrows)
{
    // wlds[p][n] = { W[2p][n], W[2p+1][n] }  (p = 0..31 k-pairs, n = 0..63)
    __shared__ v2f wlds[32 * LDS_P_STRIDE];
    float* wldsf = (float*)wlds;
    const int t = threadIdx.x;
    for (int i = t; i < 64 * 64; i += 256) {
        const int k = i >> 6, n = i & 63;
        wldsf[((k >> 1) * LDS_P_STRIDE + n) * 2 + (k & 1)] = W[i];
    }
    __syncthreads();

    const int wave = t >> 5;
    const int lane = t & 31;
    const int row0 = blockIdx.x * 128 + wave * 16;
    if (row0 >= nrows) return;

    const int lanelo = lane & 15;
    const int laneg  = lane >> 4;                 // 0: K+0,1   1: K+2,3

    int rowA = row0 + lanelo;
    if (rowA >= nrows) rowA = nrows - 1;          // clamp; masked at store
    const float* xrow = X + (size_t)rowA * 64;

    v8f acc[4];
#pragma unroll
    for (int tc = 0; tc < 4; ++tc) acc[tc] = (v8f){0,0,0,0,0,0,0,0};

#pragma unroll
    for (int k0 = 0; k0 < 64; k0 += 4) {
        const int kk = k0 + (laneg << 1);
        const v2f a = *(const v2f*)(xrow + kk);                 // b64 load
        const v2f* brow = wlds + ((k0 >> 1) + laneg) * LDS_P_STRIDE;
#pragma unroll
        for (int tc = 0; tc < 4; ++tc) {
            const v2f b = brow[tc * 16 + lanelo];               // ds_load_b64
            acc[tc] = __builtin_amdgcn_wmma_f32_16x16x4_f32(
                false, a, false, b, (short)0, acc[tc], false, false);
        }
    }

    const int mbase = row0 + (laneg << 3);        // C/D: M=r (lo) / M=r+8 (hi)
#pragma unroll
    for (int tc = 0; tc < 4; ++tc) {
        const int n = tc * 16 + lanelo;
        const float bv = bias[n];
#pragma unroll
        for (int r = 0; r < 8; ++r) {
            const int m = mbase + r;
            if (m < nrows) Hout[(size_t)m * 64 + n] = acc[tc][r] + bv;
        }
    }
}

// ---------------- small utility kernels ------------------------------------
__global__ void zero_kernel(float* __restrict__ p, int n) {
    int i = blockIdx.x * blockDim.x + threadIdx.x;
    if (i < n) p[i] = 0.0f;
}

__global__ void deg_kernel(const int* __restrict__ dst, float* __restrict__ deg,
                           int nedges) {
    int e = blockIdx.x * blockDim.x + threadIdx.x;
    if (e < nedges) atomicAdd(&deg[dst[e]], 1.0f);
}

__global__ void deg_finalize(float* __restrict__ dinv, int n) {
    int i = blockIdx.x * blockDim.x + threadIdx.x;
    if (i < n) dinv[i] = rsqrtf(dinv[i] + 1.0f);    // +1 self loop
}

// agg = h * dinv^2  (self-loop term; also the scatter init). float4 traffic.
__global__ void init_agg(const float* __restrict__ h,
                         const float* __restrict__ dinv,
                         float* __restrict__ agg, int nvec) // nvec = N*16
{
    int i = blockIdx.x * blockDim.x + threadIdx.x;
    if (i >= nvec) return;
    const float di = dinv[i >> 4];                  // 16 float4 per row
    const float d2 = di * di;
    v4f v = ((const v4f*)h)[i];
    v.x *= d2; v.y *= d2; v.z *= d2; v.w *= d2;
    ((v4f*)agg)[i] = v;
}

// 64 threads per edge, feature-contiguous: coalesced gather + fp32 L2 atomics
__global__ __launch_bounds__(256) void scatter_kernel(
    const float* __restrict__ h, const int* __restrict__ src,
    const int* __restrict__ dst, const float* __restrict__ dinv,
    float* __restrict__ agg, int nedges)
{
    const int tid = blockIdx.x * 256 + threadIdx.x;
    const int e = tid >> 6;
    const int c = tid & 63;
    if (e >= nedges) return;
    const int s = src[e];
    const int d = dst[e];
    const float coef = dinv[s] * dinv[d];
    atomicAdd(&agg[(size_t)d * 64 + c], h[(size_t)s * 64 + c] * coef);
}

// column-wise sum / sumsq partials -> stats[0..63]=sum, stats[64..127]=sumsq
__global__ __launch_bounds__(256) void stats_kernel(
    const float* __restrict__ agg, float* __restrict__ stats, int nrows)
{
    __shared__ float ssum[256];
    __shared__ float ssq[256];
    const int t = threadIdx.x;
    const int c = t & 63;
    const int rg = t >> 6;                  // 4 row-groups per block
    float s = 0.0f, q = 0.0f;
    for (int r = blockIdx.x * 4 + rg; r < nrows; r += gridDim.x * 4) {
        const float v = agg[(size_t)r * 64 + c];
        s += v; q += v * v;
    }
    ssum[t] = s; ssq[t] = q;
    __syncthreads();
    if (t < 64) {
        s = ssum[t] + ssum[t + 64] + ssum[t + 128] + ssum[t + 192];
        q = ssq[t]  + ssq[t + 64]  + ssq[t + 128]  + ssq[t + 192];
        atomicAdd(&stats[c], s);
        atomicAdd(&stats[64 + c], q);
    }
}

// normalize + affine (+ReLU), float4 traffic; 16 float4 per row, 4 cols each
__global__ void bn_kernel(const float* __restrict__ agg,
                          const float* __restrict__ stats,
                          const float* __restrict__ g,
                          const float* __restrict__ be,
                          float* __restrict__ out,
                          int nvec, float invN, int do_relu) // nvec = N*16
{
    int i = blockIdx.x * blockDim.x + threadIdx.x;
    if (i >= nvec) return;
    const int c0 = (i & 15) * 4;
    v4f mu, sc, bb;
#pragma unroll
    for (int j = 0; j < 4; ++j) {
        const int c = c0 + j;
        const float m = stats[c] * invN;
        const float var = stats[64 + c] * invN - m * m;
        mu[j] = m;
        sc[j] = rsqrtf(var + BN_EPS) * g[c];
        bb[j] = be[c];
    }
    v4f v = ((const v4f*)agg)[i];
#pragma unroll
    for (int j = 0; j < 4; ++j) {
        float r = (v[j] - mu[j]) * sc[j] + bb[j];
        if (do_relu) r = fmaxf(r, 0.0f);
        v[j] = r;
    }
    ((v4f*)out)[i] = v;
}

// ---------------- host side -------------------------------------------------
static void run_layer(const float* xin, const float* W, const float* b,
                      const float* g, const float* be,
                      const int* src, const int* dst, const float* dinv,
                      float* h, float* agg, float* stats, float* xout,
                      int N, int E, int do_relu, hipStream_t stream)
{
    const int NV = N * 16;                         // float4 elements
    const int gemm_blocks = (N + 127) / 128;
    const int vec_blocks  = (NV + 255) / 256;
    const long long sc_threads = (long long)E * 64;
    const int sc_blocks = (int)((sc_threads + 255) / 256);

    gcn_gemm_wmma<<<gemm_blocks, 256, 0, stream>>>(xin, W, b, h, N);
    init_agg<<<vec_blocks, 256, 0, stream>>>(h, dinv, agg, NV);
    scatter_kernel<<<sc_blocks, 256, 0, stream>>>(h, src, dst, dinv, agg, E);
    zero_kernel<<<1, 128, 0, stream>>>(stats, 128);
    stats_kernel<<<1024, 256, 0, stream>>>(agg, stats, N);
    bn_kernel<<<vec_blocks, 256, 0, stream>>>(agg, stats, g, be, xout, NV,
                                              1.0f / (float)N, do_relu);
}

extern "C" void kernel_launch(void* const* d_in, const int* in_sizes, int n_in,
                              void* d_out, int out_size, void* d_ws, size_t ws_size,
                              hipStream_t stream)
{
    const float* x   = (const float*)d_in[0];
    const int*   src = (const int*)  d_in[1];
    const int*   dst = (const int*)  d_in[2];
    const float* W0  = (const float*)d_in[3];
    const float* b0  = (const float*)d_in[4];
    const float* W1  = (const float*)d_in[5];
    const float* b1  = (const float*)d_in[6];
    const float* W2  = (const float*)d_in[7];
    const float* b2  = (const float*)d_in[8];
    const float* g0  = (const float*)d_in[9];
    const float* be0 = (const float*)d_in[10];
    const float* g1  = (const float*)d_in[11];
    const float* be1 = (const float*)d_in[12];
    const float* g2  = (const float*)d_in[13];
    const float* be2 = (const float*)d_in[14];

    const int N = in_sizes[0] / 64;
    const int E = in_sizes[1];
    const int NE = N * 64;

    float* ws    = (float*)d_ws;
    float* h     = ws;                   // N*64
    float* agg   = ws + (size_t)NE;      // N*64
    float* xb    = ws + (size_t)2 * NE;  // N*64
    float* dinv  = ws + (size_t)3 * NE;  // N  (degree computed in place)
    float* stats = dinv + N;             // 128

    float* out = (float*)d_out;

    // degree -> D^{-1/2}
    zero_kernel<<<(N + 255) / 256, 256, 0, stream>>>(dinv, N);
    deg_kernel<<<(E + 255) / 256, 256, 0, stream>>>(dst, dinv, E);
    deg_finalize<<<(N + 255) / 256, 256, 0, stream>>>(dinv, N);

    // layer 0: x -> xb (ReLU)
    run_layer(x,  W0, b0, g0, be0, src, dst, dinv, h, agg, stats, xb,  N, E, 1, stream);
    // layer 1: xb -> xb (ReLU)  (gemm consumes xb before bn overwrites it)
    run_layer(xb, W1, b1, g1, be1, src, dst, dinv, h, agg, stats, xb,  N, E, 1, stream);
    // layer 2: xb -> out (no ReLU)
    run_layer(xb, W2, b2, g2, be2, src, dst, dinv, h, agg, stats, out, N, E, 0, stream);
}